// DyGraph_55937654063571
// MI455X (gfx1250) — compile-verified
//
#include <hip/hip_runtime.h>
#include <hip/hip_bf16.h>

// ---------------------------------------------------------------------------
// Problem constants (from the reference)
// ---------------------------------------------------------------------------
#define SEQ      64
#define ULEN     1024
#define NTOK     (SEQ * ULEN)        // 65536
#define NLOC     200000
#define NCENT    256
#define KCAND    128
#define TOPK     10
#define HID      64
#define EMB      20
#define QPAD     12                  // padded row width for 10-wide tables (48B, 16B aligned)
#define HSTR     24                  // LDS row stride for hidden staging
#define WSLOT    (16 * HSTR + 32)    // per-wave LDS slice: 16 H rows + dump area

typedef __attribute__((ext_vector_type(2))) float v2f;
typedef __attribute__((ext_vector_type(8))) float v8f;

// fp32 WMMA:  D(16x16) += A(16x4) * B(4x16)
// A frag: lane l holds rows M = l&15, K = {2*(l>>4), 2*(l>>4)+1}
// B frag: lane l holds cols N = l&15, K = {2*(l>>4), 2*(l>>4)+1}
// C frag: lane l, vgpr r holds C[M = r + 8*(l>>4)][N = l&15]
__device__ __forceinline__ v8f wmma_f32(v2f a, v2f b, v8f c) {
    return __builtin_amdgcn_wmma_f32_16x16x4_f32(
        /*neg_a=*/false, a, /*neg_b=*/false, b,
        /*c_mod=*/(short)0, c, /*reuse_a=*/false, /*reuse_b=*/false);
}

// ---------------------------------------------------------------------------
// Stage a weight matrix into LDS, pair-interleaved and zero-padded:
//   dst[p*ncols*2 + c*2 + v] = (c < realCols) ? W[(2p+v)*realCols + c] : 0
// so a B fragment (rows k,k+1 / one column) is a single 8B ds_load_b64.
// ---------------------------------------------------------------------------
__device__ __forceinline__ void stage_pairs(float* dst, const float* __restrict__ W,
                                            int npairs, int ncols, int realCols,
                                            int tid, int nthreads)
{
    const int total = npairs * ncols * 2;
    for (int i = tid; i < total; i += nthreads) {
        const int p   = i / (ncols * 2);
        const int rem = i - p * (ncols * 2);
        const int c   = rem >> 1;
        const int v   = rem & 1;
        dst[i] = (c < realCols) ? W[(2 * p + v) * realCols + c] : 0.0f;
    }
}

// zero-padded bias staging: b1 -> 32 floats, b2 -> 16 floats
__device__ __forceinline__ void stage_bias(float* b1l, const float* __restrict__ b1,
                                           float* b2l, const float* __restrict__ b2,
                                           int tid)
{
    if (tid < 32)                b1l[tid]      = (tid < EMB)       ? b1[tid]      : 0.0f;
    else if (tid < 48)           b2l[tid - 32] = (tid - 32 < TOPK) ? b2[tid - 32] : 0.0f;
}

// ---------------------------------------------------------------------------
// Shared tail: relu(C + b1) staged through per-wave LDS slice, then
// H[16x20] @ W2[20x10] via WMMA (+b2), stored padded to QPAD floats per row.
// All predication is done with address selects -> EXEC stays full throughout.
// ---------------------------------------------------------------------------
__device__ __forceinline__ void hidden_to_out(
    float* hrow, const float* W2l, const float* b1l, const float* b2l,
    v8f c0, v8f c1, float* __restrict__ dst, int row0, int lane)
{
    const int m = lane & 15;
    const int h = lane >> 4;

    const float bm0 = b1l[m];          // cols 0..15
    const float bm1 = b1l[16 + m];     // cols 16..31 (zero padded)
    const float b2v = b2l[(m < TOPK) ? m : 0];

    // scatter relu(H) into LDS; lanes whose col-16+ slot doesn't exist (m>=4)
    // dump into a per-wave scratch word instead of taking a divergent branch.
    const int dumpoff = 16 * HSTR + lane;
    #pragma unroll
    for (int r = 0; r < 8; ++r) {
        const int row = r + 8 * h;
        hrow[row * HSTR + m] = fmaxf(c0[r] + bm0, 0.0f);
        const int a1 = (m < EMB - 16) ? (row * HSTR + 16 + m) : dumpoff;
        hrow[a1] = fmaxf(c1[r] + bm1, 0.0f);
    }
    // wave-local LDS RAW: DS ops of one wave are in-order; just drain DScnt.
    asm volatile("s_wait_dscnt 0x0" ::: "memory");

    v8f d = {};
    #pragma unroll
    for (int c = 0; c < 5; ++c) {                 // K = 20, 4 per step
        const int kb = 4 * c + 2 * h;             // pair index = 2c + h
        v2f a = *(const v2f*)&hrow[m * HSTR + kb];
        v2f b = *(const v2f*)&W2l[((2 * c + h) * 16 + m) * 2];
        d = wmma_f32(a, b, d);
    }

    // lanes m>=10 write the (unread) padding column 10 -> branchless stores
    const int colw = (m < TOPK) ? m : TOPK;
    #pragma unroll
    for (int r = 0; r < 8; ++r) {
        const int row = r + 8 * h;
        dst[(row0 + row) * QPAD + colw] = d[r] + b2v;
    }
}

// ---------------------------------------------------------------------------
// Kernel A: x_embedding_dy = relu(vecs @ Wv1 + bv1) @ Wv2 + bv2   [NLOC,10]
// One wave per 16-row tile; 256-thread blocks = 8 tiles per block.
// ---------------------------------------------------------------------------
__global__ void __launch_bounds__(256)
vec_mlp_wmma(const float* __restrict__ vecs,
             const float* __restrict__ W1, const float* __restrict__ b1,
             const float* __restrict__ W2, const float* __restrict__ b2,
             float* __restrict__ dy)
{
    __shared__ float W1l[10 * 32 * 2];            // 20x20 padded to 20x32, paired
    __shared__ float W2l[10 * 16 * 2];            // 20x10 padded to 20x16, paired
    __shared__ float b1l[32];
    __shared__ float b2l[16];
    __shared__ float Hlds[8 * WSLOT];
    stage_pairs(W1l, W1, 10, 32, EMB, threadIdx.x, 256);
    stage_pairs(W2l, W2, 10, 16, TOPK, threadIdx.x, 256);
    stage_bias(b1l, b1, b2l, b2, threadIdx.x);
    __syncthreads();

    const int wave = threadIdx.x >> 5;
    const int lane = threadIdx.x & 31;
    const int tile = blockIdx.x * 8 + wave;       // wave-uniform
    if (tile < NLOC / 16) {                       // uniform branch: EXEC stays full
        const int m  = lane & 15;
        const int h  = lane >> 4;
        const int r0 = tile * 16;
        const float* vp = vecs + (r0 + m) * EMB;

        v8f c0 = {}, c1 = {};
        #pragma unroll
        for (int c = 0; c < 5; ++c) {             // K = 20
            const int kb = 4 * c + 2 * h;
            v2f a = *(const v2f*)(vp + kb);       // 8B-aligned global b64
            const float* wp = &W1l[((2 * c + h) * 32 + m) * 2];
            v2f b0 = *(const v2f*)(wp);           // cols 0..15
            v2f bq = *(const v2f*)(wp + 32);      // cols 16..31 (zero padded)
            c0 = wmma_f32(a, b0, c0);
            c1 = wmma_f32(a, bq, c1);
        }
        hidden_to_out(&Hlds[wave * WSLOT], W2l, b1l, b2l, c0, c1, dy, r0, lane);
    }
}

// ---------------------------------------------------------------------------
// Kernel B: q = relu(concat(shift4..shift0 embeddings) @ Ws1 + bs1) @ Ws2 + bs2
// concat is [NTOK,100]; one wave per 16-token tile (tokens in a tile share s).
// ---------------------------------------------------------------------------
__global__ void __launch_bounds__(256)
seq_mlp_wmma(const int* __restrict__ x, const float* __restrict__ vecs,
             const float* __restrict__ W1, const float* __restrict__ b1,
             const float* __restrict__ W2, const float* __restrict__ b2,
             float* __restrict__ q)
{
    __shared__ float W1l[50 * 32 * 2];            // 100x20 padded to 100x32, paired
    __shared__ float W2l[10 * 16 * 2];
    __shared__ float b1l[32];
    __shared__ float b2l[16];
    __shared__ float Hlds[8 * WSLOT];
    stage_pairs(W1l, W1, 50, 32, EMB, threadIdx.x, 256);
    stage_pairs(W2l, W2, 10, 16, TOPK, threadIdx.x, 256);
    stage_bias(b1l, b1, b2l, b2, threadIdx.x);
    __syncthreads();

    const int wave = threadIdx.x >> 5;
    const int lane = threadIdx.x & 31;
    const int tile = blockIdx.x * 8 + wave;       // 4096 tiles exactly
    const int m  = lane & 15;
    const int h  = lane >> 4;
    const int n0 = tile * 16;
    const int s  = n0 >> 10;                      // ULEN = 1024
    const int u  = (n0 & (ULEN - 1)) + m;

    // 5 gathered location ids per token: groups g=0..4 are shift 4,3,2,1,0
    int idxs[5];
    #pragma unroll
    for (int g = 0; g < 5; ++g) {
        const int k  = 4 - g;
        const int sp = (s < k) ? s : (s - k);
        idxs[g] = x[sp * ULEN + u];
    }

    v8f c0 = {}, c1 = {};
    #pragma unroll 5
    for (int c = 0; c < 25; ++c) {                // K = 100
        const int g  = c / 5;
        const int db = (c - g * 5) * 4 + 2 * h;   // dim within the 20-wide group
        v2f a = *(const v2f*)(vecs + idxs[g] * EMB + db);   // 8B-aligned gather
        const float* wp = &W1l[((2 * c + h) * 32 + m) * 2];
        v2f b0 = *(const v2f*)(wp);
        v2f bq = *(const v2f*)(wp + 32);
        c0 = wmma_f32(a, b0, c0);
        c1 = wmma_f32(a, bq, c1);
    }
    hidden_to_out(&Hlds[wave * WSLOT], W2l, b1l, b2l, c0, c1, q, n0, lane);
}

// ---------------------------------------------------------------------------
// Kernel C: per-token 128-candidate distance + exp score + stable top-10.
// q and dy rows are QPAD floats -> three b128 loads each. Prefetch 8 ahead.
// ---------------------------------------------------------------------------
__global__ void __launch_bounds__(256)
topk_kernel(const int* __restrict__ x, const int* __restrict__ I,
            const int* __restrict__ cand, const float* __restrict__ q,
            const float* __restrict__ dy,
            float* __restrict__ topv, int* __restrict__ topi)
{
    const int n = blockIdx.x * blockDim.x + threadIdx.x;   // 65536 threads
    const int v = x[n];
    const float* qr = q + n * QPAD;
    const float4 qa = *(const float4*)(qr);
    const float4 qb = *(const float4*)(qr + 4);
    const float4 qc = *(const float4*)(qr + 8);

    const int* cr = cand + I[v] * KCAND;

    float tv[TOPK];
    int   ti[TOPK];
    #pragma unroll
    for (int j = 0; j < TOPK; ++j) { tv[j] = -1.0f; ti[j] = 0; }

    for (int k = 0; k < KCAND; ++k) {
        const int cid = cr[k];
        // pull candidate row k+8 from L2 toward the WGP (global_prefetch_b8)
        const int kp = (k + 8 < KCAND) ? (k + 8) : (KCAND - 1);
        __builtin_prefetch(dy + cr[kp] * QPAD, 0, 0);

        const float* er = dy + cid * QPAD;
        const float4 e0 = *(const float4*)(er);
        const float4 e1 = *(const float4*)(er + 4);
        const float4 e2 = *(const float4*)(er + 8);
        float t, d2;
        t = qa.x - e0.x; d2  = t * t;
        t = qa.y - e0.y; d2 = fmaf(t, t, d2);
        t = qa.z - e0.z; d2 = fmaf(t, t, d2);
        t = qa.w - e0.w; d2 = fmaf(t, t, d2);
        t = qb.x - e1.x; d2 = fmaf(t, t, d2);
        t = qb.y - e1.y; d2 = fmaf(t, t, d2);
        t = qb.z - e1.z; d2 = fmaf(t, t, d2);
        t = qb.w - e1.w; d2 = fmaf(t, t, d2);
        t = qc.x - e2.x; d2 = fmaf(t, t, d2);
        t = qc.y - e2.y; d2 = fmaf(t, t, d2);
        float sv = __expf(-0.02f * sqrtf(d2));
        int   si = cid;
        // stable descending insertion (strict > keeps earlier index on ties,
        // matching jax.lax.top_k)
        #pragma unroll
        for (int j = 0; j < TOPK; ++j) {
            const bool  p  = sv > tv[j];
            const float t0 = tv[j];
            const int   t1 = ti[j];
            tv[j] = p ? sv : t0;  ti[j] = p ? si : t1;
            sv    = p ? t0 : sv;  si    = p ? t1 : si;
        }
    }
    #pragma unroll
    for (int j = 0; j < TOPK; ++j) {
        topv[n * TOPK + j] = tv[j];
        topi[n * TOPK + j] = ti[j];
    }
}

// ---------------------------------------------------------------------------
// Kernel D: softmax([values,1]) weighted sum of gathered 64-dim rows.
// One wave per token, 2 output dims per lane (coalesced 64-float row reads).
// ---------------------------------------------------------------------------
__global__ void __launch_bounds__(256)
combine_kernel(const int* __restrict__ x, const float* __restrict__ xen,
               const float* __restrict__ topv, const int* __restrict__ topi,
               float* __restrict__ out)
{
    const int token = blockIdx.x * 8 + (threadIdx.x >> 5);
    const int lane  = threadIdx.x & 31;
    const float E1  = 2.7182818284590452f;        // exp(1) for the self slot

    const float* tvp = topv + token * TOPK;
    float wv[TOPK];
    float wsum = E1;
    #pragma unroll
    for (int j = 0; j < TOPK; ++j) { wv[j] = __expf(tvp[j]); wsum += wv[j]; }
    const float inv = 1.0f / wsum;

    float a0 = 0.0f, a1 = 0.0f;
    #pragma unroll
    for (int j = 0; j < TOPK; ++j) {
        const float* e = xen + topi[token * TOPK + j] * HID;
        const float  w = wv[j] * inv;
        a0 = fmaf(w, e[lane], a0);
        a1 = fmaf(w, e[lane + 32], a1);
    }
    const float* e = xen + x[token] * HID;
    const float  w = E1 * inv;
    a0 = fmaf(w, e[lane], a0);
    a1 = fmaf(w, e[lane + 32], a1);

    out[token * HID + lane]      = a0;
    out[token * HID + lane + 32] = a1;
}

// ---------------------------------------------------------------------------
// Launch
// ---------------------------------------------------------------------------
extern "C" void kernel_launch(void* const* d_in, const int* in_sizes, int n_in,
                              void* d_out, int out_size, void* d_ws, size_t ws_size,
                              hipStream_t stream) {
    const int*   x     = (const int*)d_in[0];
    const int*   I     = (const int*)d_in[1];
    const int*   cand  = (const int*)d_in[2];
    const float* vecs  = (const float*)d_in[3];
    const float* xen   = (const float*)d_in[4];
    const float* Ws1   = (const float*)d_in[5];
    const float* bs1   = (const float*)d_in[6];
    const float* Ws2   = (const float*)d_in[7];
    const float* bs2   = (const float*)d_in[8];
    const float* Wv1   = (const float*)d_in[9];
    const float* bv1   = (const float*)d_in[10];
    const float* Wv2   = (const float*)d_in[11];
    const float* bv2   = (const float*)d_in[12];
    float* out = (float*)d_out;

    // workspace carve-up (~18 MB)
    float* dy   = (float*)d_ws;                       // [NLOC  x QPAD]
    float* q    = dy + (size_t)NLOC * QPAD;           // [NTOK  x QPAD]
    float* topv = q  + (size_t)NTOK * QPAD;           // [NTOK  x TOPK]
    int*   topi = (int*)(topv + (size_t)NTOK * TOPK); // [NTOK  x TOPK]

    // A: vec MLP over all locations (12500 tiles, 8 waves/block)
    vec_mlp_wmma<<<(NLOC / 16 + 7) / 8, 256, 0, stream>>>(vecs, Wv1, bv1, Wv2, bv2, dy);
    // B: query MLP over tokens (4096 tiles)
    seq_mlp_wmma<<<(NTOK / 16) / 8, 256, 0, stream>>>(x, vecs, Ws1, bs1, Ws2, bs2, q);
    // C: candidate scores + top-10
    topk_kernel<<<NTOK / 256, 256, 0, stream>>>(x, I, cand, q, dy, topv, topi);
    // D: softmax combine -> [NTOK, 64]
    combine_kernel<<<NTOK / 8, 256, 0, stream>>>(x, xen, topv, topi, out);
}